// quantizer_89833535963912
// MI455X (gfx1250) — compile-verified
//
#include <hip/hip_runtime.h>

typedef __attribute__((ext_vector_type(2))) float v2f;
typedef __attribute__((ext_vector_type(8))) float v8f;

#define NEG_TEMP_LOG2E (-14.426950408889634f) // -10 * log2(e): exp(-10*t) = exp2(K*t)

// Main kernel: one element per lane, two 16x16 WMMA distance tiles per wave.
// Requires n multiple of 256 (full waves -> EXEC all ones for WMMA).
__global__ __launch_bounds__(256) void soft_quant_wmma(
    const float* __restrict__ x, const float* __restrict__ center,
    float* __restrict__ out)
{
    const int gid  = blockIdx.x * 256 + threadIdx.x;
    const int lane = threadIdx.x & 31;
    const bool hi  = lane >= 16;

    const float xv = x[gid];

    // ---- Exact hard-quantize: replicate reference rounding & tie-breaking.
    // centers read with uniform indices -> SGPRs; strict '<' keeps first min.
    float c0   = center[0];
    float best = fabsf(xv - c0);
    float bc   = c0;
#pragma unroll
    for (int m = 1; m < 16; ++m) {
        float cm = center[m];
        float ad = fabsf(xv - cm);
        bool  lt = ad < best;
        best = lt ? ad : best;
        bc   = lt ? cm : bc;
    }
    const float d2min = best * best;   // max-logit shift for stable softmax

    // ---- A matrix (wave-constant): row m = [-2*c_m, c_m^2, 1, 0]
    // f32 16x4 A layout: lanes 0-15 hold K=0,1 (V0,V1); lanes 16-31 hold K=2,3.
    const float cml = center[lane & 15];
    v2f A;
    A.x = hi ? 1.0f : (-2.0f * cml);
    A.y = hi ? 0.0f : (cml * cml);

    // Per-lane centers for the weighted sum: D row r maps to center[r + 8*hi].
    float cme[8];
    {
        const float4* c4 = (const float4*)center;
        float4 u0 = c4[hi ? 2 : 0];
        float4 u1 = c4[hi ? 3 : 1];
        cme[0] = u0.x; cme[1] = u0.y; cme[2] = u0.z; cme[3] = u0.w;
        cme[4] = u1.x; cme[5] = u1.y; cme[6] = u1.z; cme[7] = u1.w;
    }

    // ---- B matrices: col j = [x_j, 1, x_j^2, 0] for the two 16-element tiles.
    // f32 4x16 B layout mirrors A: lanes 0-15 hold K=0,1; lanes 16-31 hold K=2,3.
    const float xs    = __shfl_xor(xv, 16);
    const float xlow  = hi ? xs : xv;   // x of element (lane&15)       -> tile 0
    const float xhigh = hi ? xv : xs;   // x of element 16+(lane&15)    -> tile 1
    v2f B0, B1;
    B0.x = hi ? (xlow  * xlow ) : xlow;   B0.y = hi ? 0.0f : 1.0f;
    B1.x = hi ? (xhigh * xhigh) : xhigh;  B1.y = hi ? 0.0f : 1.0f;

    // D[m][j] = (x_j - c_m)^2 ; D on lane l: column j=l&15, rows r+8*hi.
    v8f cz = {};
    v8f D0 = __builtin_amdgcn_wmma_f32_16x16x4_f32(false, A, false, B0,
                                                   (short)0, cz, false, false);
    v8f D1 = __builtin_amdgcn_wmma_f32_16x16x4_f32(false, A, false, B1,
                                                   (short)0, cz, false, false);

    // d2min of each tile's element (partner's via xor-16).
    const float d2min_s  = __shfl_xor(d2min, 16);
    const float d2min_t0 = hi ? d2min_s : d2min;
    const float d2min_t1 = hi ? d2min   : d2min_s;
    const float sh0 = NEG_TEMP_LOG2E * d2min_t0;  // fold shift into one FMA
    const float sh1 = NEG_TEMP_LOG2E * d2min_t1;

    // exp over this lane's 16 entries (8 per tile) + partial sums.
    float num0 = 0.0f, den0 = 0.0f, num1 = 0.0f, den1 = 0.0f;
#pragma unroll
    for (int r = 0; r < 8; ++r) {
        float a0 = fminf(fmaf(NEG_TEMP_LOG2E, D0[r], -sh0), 0.0f); // clamp<=0: NaN-proof
        float e0 = __builtin_amdgcn_exp2f(a0);                     // raw v_exp_f32
        den0 += e0;  num0 = fmaf(e0, cme[r], num0);
        float a1 = fminf(fmaf(NEG_TEMP_LOG2E, D1[r], -sh1), 0.0f);
        float e1 = __builtin_amdgcn_exp2f(a1);
        den1 += e1;  num1 = fmaf(e1, cme[r], num1);
    }

    // Exchange partner-element partials: lane l owns tile0 col l, lane l+16 owns tile1 col l.
    const float sendn = hi ? num0 : num1;
    const float sendd = hi ? den0 : den1;
    const float recvn = __shfl_xor(sendn, 16);
    const float recvd = __shfl_xor(sendd, 16);
    float mynum = (hi ? num1 : num0) + recvn;
    float myden = (hi ? den1 : den0) + recvd;

    myden = fmaxf(myden, 1e-30f);          // guard (den >= ~1 when healthy)
    const float soft = __fdividef(mynum, myden);

    // Straight-through estimator forward: (hard - soft) + soft, same rounding as ref.
    out[gid] = (bc - soft) + soft;
}

// Scalar tail kernel for any ragged remainder (not hit for the 16.7M case).
__global__ void soft_quant_tail(const float* __restrict__ x,
                                const float* __restrict__ center,
                                float* __restrict__ out, int start, int n)
{
    int gid = start + blockIdx.x * 256 + threadIdx.x;
    if (gid >= n) return;
    float xv = x[gid];
    float c0 = center[0];
    float best = fabsf(xv - c0);
    float bc = c0;
    float d[16];
    d[0] = xv - c0;
#pragma unroll
    for (int m = 1; m < 16; ++m) {
        float cm = center[m];
        d[m] = xv - cm;
        float ad = fabsf(d[m]);
        bool lt = ad < best;
        best = lt ? ad : best;
        bc   = lt ? cm : bc;
    }
    float d2min = best * best;
    float sh = NEG_TEMP_LOG2E * d2min;
    float num = 0.0f, den = 0.0f;
#pragma unroll
    for (int m = 0; m < 16; ++m) {
        float d2 = d[m] * d[m];
        float a  = fminf(fmaf(NEG_TEMP_LOG2E, d2, -sh), 0.0f);
        float e  = __builtin_amdgcn_exp2f(a);
        den += e;
        num  = fmaf(e, center[m], num);
    }
    float soft = __fdividef(num, fmaxf(den, 1e-30f));
    out[gid] = (bc - soft) + soft;
}

extern "C" void kernel_launch(void* const* d_in, const int* in_sizes, int n_in,
                              void* d_out, int out_size, void* d_ws, size_t ws_size,
                              hipStream_t stream) {
    const float* x      = (const float*)d_in[0];
    const float* center = (const float*)d_in[1];
    float*       out    = (float*)d_out;
    const int n = in_sizes[0];            // 16*64*128*128 = 16,777,216

    const int nmain = n & ~255;           // full 256-thread blocks, full waves
    if (nmain > 0) {
        soft_quant_wmma<<<nmain / 256, 256, 0, stream>>>(x, center, out);
    }
    const int rem = n - nmain;
    if (rem > 0) {
        soft_quant_tail<<<(rem + 255) / 256, 256, 0, stream>>>(x, center, out, nmain, n);
    }
}